// TSAttention_71571335020796
// MI455X (gfx1250) — compile-verified
//
#include <hip/hip_runtime.h>

// Problem constants (from reference)
constexpr int NS  = 256;
constexpr int B_SZ = 4;
constexpr int DM  = 512;
constexpr int NH  = 8;
constexpr int DH  = 64;
constexpr int NHD = NH * DH; // 512

typedef __attribute__((ext_vector_type(2))) float v2f;
typedef __attribute__((ext_vector_type(8))) float v8f;

// D = A(16x4,f32) * B(4x16,f32) + C(16x16,f32), exact f32 WMMA.
// Lane layout (wave32): A: m=lane&15, k={2*(lane>>4), 2*(lane>>4)+1}
//                       B: n=lane&15, same k pair
//                       C/D: VGPR i -> row (i + 8*(lane>>4)), col lane&15
__device__ __forceinline__ v8f wmma_f32_k4(v2f a, v2f b, v8f c) {
  return __builtin_amdgcn_wmma_f32_16x16x4_f32(false, a, false, b, (short)0, c,
                                               false, false);
}

// ---------------------------------------------------------------------------
// K1: projection GEMM  C[r][c] = sum_d X[r][d]*W[c][d] + bias[c]
// X: (1024 x 512) row-major, row r = node*B + b ; W: (512 x 512) row-major
// Scatter store into [b][h][node][e] layout (c = h*DH + e).
// Grid: (64, 32) tiles of 16x16, block = 32 (one wave).
// ---------------------------------------------------------------------------
__global__ void proj_kernel(const float* __restrict__ X,
                            const float* __restrict__ W,
                            const float* __restrict__ bias,
                            float* __restrict__ outp) {
  const int lane = threadIdx.x;
  const int half = lane >> 4;
  const int lm   = lane & 15;
  const int m0 = blockIdx.x * 16;  // row base (0..1023)
  const int n0 = blockIdx.y * 16;  // col base (0..511)

  const float* xrow = X + (m0 + lm) * DM;
  const float* wrow = W + (n0 + lm) * DM;

  v8f acc = {};
#pragma unroll 4
  for (int k0 = 0; k0 < DM; k0 += 4) {
    v2f a = *(const v2f*)(xrow + k0 + 2 * half);
    v2f b = *(const v2f*)(wrow + k0 + 2 * half);
    acc = wmma_f32_k4(a, b, acc);
  }

  const int c    = n0 + lm;
  const float bi = bias[c];
  const int h = c >> 6, e = c & 63;
#pragma unroll
  for (int i = 0; i < 8; ++i) {
    const int r    = m0 + i + 8 * half;  // r = node*B + b
    const int node = r >> 2;
    const int bb   = r & 3;
    outp[(((bb * NH + h) * NS) + node) * DH + e] = acc[i] + bi;
  }
}

// ---------------------------------------------------------------------------
// K2: A2[b] = clip(A[b] @ A[b], 0, 1)   (B x 256 x 256)
// Grid: (16, 16, B), block = 32.
// ---------------------------------------------------------------------------
__global__ void adj2_kernel(const float* __restrict__ A,
                            float* __restrict__ A2) {
  const int lane = threadIdx.x;
  const int half = lane >> 4;
  const int lm   = lane & 15;
  const int q0 = blockIdx.x * 16;
  const int n0 = blockIdx.y * 16;
  const int b  = blockIdx.z;
  const float* Ab = A + b * NS * NS;

  v8f acc = {};
#pragma unroll 4
  for (int k0 = 0; k0 < NS; k0 += 4) {
    v2f a = *(const v2f*)(Ab + (q0 + lm) * NS + k0 + 2 * half);
    v2f bb;
    bb.x = Ab[(k0 + 2 * half + 0) * NS + n0 + lm];
    bb.y = Ab[(k0 + 2 * half + 1) * NS + n0 + lm];
    acc = wmma_f32_k4(a, bb, acc);
  }
#pragma unroll
  for (int i = 0; i < 8; ++i) {
    const int q = q0 + i + 8 * half;
    A2[(b * NS + q) * NS + n0 + lm] = fminf(fmaxf(acc[i], 0.0f), 1.0f);
  }
}

// ---------------------------------------------------------------------------
// K3: pairwise additive-attention scores + adjacency bias + mask.
// score(b,h,q,k) = 0.505*sum_e s_e*(q_e+k_e) + 0.495*sum_e s_e*|q_e+k_e|
// Grid: (16 q-tiles, NH, B), block = 256. LDS rows padded to 65 floats.
// ---------------------------------------------------------------------------
__global__ void attn_kernel(const float* __restrict__ qp,
                            const float* __restrict__ kp,
                            const float* __restrict__ sproj,
                            const float* __restrict__ A,
                            const float* __restrict__ A2,
                            const float* __restrict__ nw,
                            const float* __restrict__ amask,
                            float* __restrict__ attn) {
  __shared__ float sQ[16 * 65];
  __shared__ float sK[64 * 65];
  __shared__ float sS[DH];

  const int t  = threadIdx.x;  // 0..255
  const int q0 = blockIdx.x * 16;
  const int h  = blockIdx.y;
  const int b  = blockIdx.z;

  const float* qbh = qp + ((b * NH + h) * NS) * DH;
  const float* kbh = kp + ((b * NH + h) * NS) * DH;

  if (t < DH) sS[t] = sproj[h * DH + t];
  for (int i = t; i < 16 * DH; i += 256)
    sQ[(i >> 6) * 65 + (i & 63)] = qbh[q0 * DH + i];

  const int ql = t >> 4;
  const int kl = t & 15;
  const int qg = q0 + ql;
  const float nw0 = nw[0 * NH + h];
  const float nw1 = nw[1 * NH + h];
  const float nw2 = nw[2 * NH + h];
  const float* qrow = &sQ[ql * 65];

  for (int c0 = 0; c0 < NS; c0 += 64) {
    __syncthreads();
    for (int i = t; i < 64 * DH; i += 256)
      sK[(i >> 6) * 65 + (i & 63)] = kbh[c0 * DH + i];
    __syncthreads();
#pragma unroll
    for (int j = 0; j < 4; ++j) {
      const int kc = kl + 16 * j;   // k within chunk
      const int kg = c0 + kc;       // global k
      const float* krow = &sK[kc * 65];
      float accL = 0.0f, accA = 0.0f;
#pragma unroll 8
      for (int e = 0; e < DH; ++e) {
        const float tt = qrow[e] + krow[e];
        const float s  = sS[e];
        accL = fmaf(s, tt, accL);
        accA = fmaf(s, fabsf(tt), accA);
      }
      const float score = 0.505f * accL + 0.495f * accA;
      const float av  = A[(b * NS + qg) * NS + kg];
      const float a2v = A2[(b * NS + qg) * NS + kg];
      const float eye = (qg == kg) ? 1.0f : 0.0f;
      float val;
      if (eye + av + a2v == 0.0f)
        val = -__builtin_inff();
      else
        val = score + nw0 * eye + nw1 * av + nw2 * a2v + amask[qg * NS + kg];
      attn[((b * NH + h) * NS + qg) * NS + kg] = val;
    }
  }
}

// ---------------------------------------------------------------------------
// K4: row-wise softmax over last dim (256). One block of 256 per row.
// ---------------------------------------------------------------------------
__global__ void softmax_kernel(float* __restrict__ attn) {
  __shared__ float red[256];
  const int t = threadIdx.x;
  float* row = attn + (size_t)blockIdx.x * NS;
  const float v = row[t];
  red[t] = v;
  __syncthreads();
  for (int s = 128; s > 0; s >>= 1) {
    if (t < s) red[t] = fmaxf(red[t], red[t + s]);
    __syncthreads();
  }
  const float mx = red[0];
  __syncthreads();
  const float ex = expf(v - mx);  // v == -inf -> 0 (diag keeps mx finite)
  red[t] = ex;
  __syncthreads();
  for (int s = 128; s > 0; s >>= 1) {
    if (t < s) red[t] += red[t + s];
    __syncthreads();
  }
  row[t] = ex / red[0];
}

// ---------------------------------------------------------------------------
// K5: out_pre = relu( attn(b,h) @ kp(b,h) )  per (b,h): (256x256)@(256x64)
// Store to obuf row-major (1024 x 512), row = q*B + b, col = h*DH + d.
// Grid: (16 q-tiles, 4 d-tiles, B*NH), block = 32.
// ---------------------------------------------------------------------------
__global__ void av_kernel(const float* __restrict__ attn,
                          const float* __restrict__ kp,
                          float* __restrict__ obuf) {
  const int lane = threadIdx.x;
  const int half = lane >> 4;
  const int lm   = lane & 15;
  const int q0 = blockIdx.x * 16;
  const int d0 = blockIdx.y * 16;
  const int bh = blockIdx.z;
  const int b = bh / NH, h = bh % NH;

  const float* Ab = attn + (size_t)bh * NS * NS;
  const float* Kb = kp + (size_t)bh * NS * DH;

  v8f acc = {};
#pragma unroll 4
  for (int k0 = 0; k0 < NS; k0 += 4) {
    v2f a = *(const v2f*)(Ab + (q0 + lm) * NS + k0 + 2 * half);
    v2f bb;
    bb.x = Kb[(k0 + 2 * half + 0) * DH + d0 + lm];
    bb.y = Kb[(k0 + 2 * half + 1) * DH + d0 + lm];
    acc = wmma_f32_k4(a, bb, acc);
  }
#pragma unroll
  for (int i = 0; i < 8; ++i) {
    const int q = q0 + i + 8 * half;
    obuf[(q * B_SZ + b) * NHD + h * DH + d0 + lm] = fmaxf(acc[i], 0.0f);
  }
}

// ---------------------------------------------------------------------------
// K6: final projection  out[r][m] = sum_c obuf[r][c]*Wo[m][c] + bo[m]
// Grid: (64, 32), block = 32.
// ---------------------------------------------------------------------------
__global__ void outproj_kernel(const float* __restrict__ X,
                               const float* __restrict__ W,
                               const float* __restrict__ bias,
                               float* __restrict__ out) {
  const int lane = threadIdx.x;
  const int half = lane >> 4;
  const int lm   = lane & 15;
  const int m0 = blockIdx.x * 16;
  const int n0 = blockIdx.y * 16;

  const float* xrow = X + (m0 + lm) * NHD;
  const float* wrow = W + (n0 + lm) * NHD;

  v8f acc = {};
#pragma unroll 4
  for (int k0 = 0; k0 < NHD; k0 += 4) {
    v2f a = *(const v2f*)(xrow + k0 + 2 * half);
    v2f b = *(const v2f*)(wrow + k0 + 2 * half);
    acc = wmma_f32_k4(a, b, acc);
  }
  const float bi = bias[n0 + lm];
#pragma unroll
  for (int i = 0; i < 8; ++i) {
    const int r = m0 + i + 8 * half;
    out[r * DM + n0 + lm] = acc[i] + bi;
  }
}

// ---------------------------------------------------------------------------
extern "C" void kernel_launch(void* const* d_in, const int* in_sizes, int n_in,
                              void* d_out, int out_size, void* d_ws,
                              size_t ws_size, hipStream_t stream) {
  const float* query = (const float*)d_in[0];
  // d_in[1] (key) and d_in[2] (value) are unused by the reference.
  const float* adj   = (const float*)d_in[3];
  const float* amask = (const float*)d_in[4];
  const float* Wq    = (const float*)d_in[5];
  const float* bq    = (const float*)d_in[6];
  const float* Wk    = (const float*)d_in[7];
  const float* bk    = (const float*)d_in[8];
  const float* sproj = (const float*)d_in[9];
  const float* nw    = (const float*)d_in[10];
  const float* Wo    = (const float*)d_in[11];
  const float* bo    = (const float*)d_in[12];
  float* out = (float*)d_out;

  // Workspace carve-up (floats): ~15 MB total.
  float* ws   = (float*)d_ws;
  float* qp   = ws;                                  // B*NH*NS*DH = 524288
  float* kp   = qp + (size_t)B_SZ * NH * NS * DH;    // 524288
  float* A2   = kp + (size_t)B_SZ * NH * NS * DH;    // B*NS*NS   = 262144
  float* attn = A2 + (size_t)B_SZ * NS * NS;         // B*NH*NS*NS= 2097152
  float* obuf = attn + (size_t)B_SZ * NH * NS * NS;  // 524288

  proj_kernel<<<dim3(64, 32), 32, 0, stream>>>(query, Wq, bq, qp);
  proj_kernel<<<dim3(64, 32), 32, 0, stream>>>(query, Wk, bk, kp);
  adj2_kernel<<<dim3(16, 16, B_SZ), 32, 0, stream>>>(adj, A2);
  attn_kernel<<<dim3(16, NH, B_SZ), 256, 0, stream>>>(qp, kp, sproj, adj, A2,
                                                      nw, amask, attn);
  softmax_kernel<<<B_SZ * NH * NS, NS, 0, stream>>>(attn);
  av_kernel<<<dim3(16, 4, B_SZ * NH), 32, 0, stream>>>(attn, kp, obuf);
  outproj_kernel<<<dim3(64, 32), 32, 0, stream>>>(obuf, Wo, bo, out);
}